// GCNlayer_58583353917609
// MI455X (gfx1250) — compile-verified
//
#include <hip/hip_runtime.h>
#include <math.h>

typedef __attribute__((ext_vector_type(2))) float v2f;
typedef __attribute__((ext_vector_type(8))) float v8f;

#define NN 8192
#define DD 128
#define KC 64   // K-chunk staged in LDS per iteration

// ---------------------------------------------------------------------------
// Kernel 1: deg[n] = sum_m A[n,m];  dinv[n] = rsqrt(deg[n])
// ---------------------------------------------------------------------------
__global__ void __launch_bounds__(256) k_degree(const float* __restrict__ A,
                                                float* __restrict__ dinv) {
    __shared__ float red[256];
    const int row = blockIdx.x;
    const float4* a4 = (const float4*)(A + (size_t)row * NN);
    float s = 0.f;
    for (int i = threadIdx.x; i < NN / 4; i += 256) {
        float4 v = a4[i];
        s += v.x + v.y + v.z + v.w;
    }
    red[threadIdx.x] = s;
    __syncthreads();
    for (int off = 128; off > 0; off >>= 1) {
        if (threadIdx.x < off) red[threadIdx.x] += red[threadIdx.x + off];
        __syncthreads();
    }
    if (threadIdx.x == 0) dinv[row] = rsqrtf(red[0]);
}

// ---------------------------------------------------------------------------
// Kernel 2: Hs[m,o] = dinv[m] * sum_d X[m,d] * W[o,d]   (column scale fused)
// ---------------------------------------------------------------------------
__global__ void __launch_bounds__(128) k_xw(const float* __restrict__ X,
                                            const float* __restrict__ W,
                                            const float* __restrict__ dinv,
                                            float* __restrict__ Hs) {
    __shared__ float xrow[DD];
    const int row = blockIdx.x;
    const int o = threadIdx.x;
    xrow[o] = X[(size_t)row * DD + o];
    __syncthreads();
    const float* w = W + (size_t)o * DD;
    float acc = 0.f;
#pragma unroll 8
    for (int d = 0; d < DD; ++d) acc += xrow[d] * w[d];
    Hs[(size_t)row * DD + o] = acc * dinv[row];
}

// ---------------------------------------------------------------------------
// Kernel 3: out = relu( dinv[n] * (A @ Hs) )  via V_WMMA_F32_16X16X4_F32
// Block: 256 threads = 8 wave32s; output tile = 16 rows x 128 cols.
// A tile (16 x KC fp32) staged via global_load_async_to_lds_b128 into a
// double-buffered LDS tile so the DMA of chunk k+1 overlaps the 16-WMMA
// chain on chunk k. ASYNCcnt + one barrier per iteration.
// ---------------------------------------------------------------------------
__global__ void __launch_bounds__(256) k_spmm(const float* __restrict__ A,
                                              const float* __restrict__ Hs,
                                              const float* __restrict__ dinv,
                                              float* __restrict__ out) {
    // row padded to 68 floats: LDS bank = (4*l + k) % 64 -> conflict-free for
    // the 16-lane A-fragment reads; 68*4 = 272 B keeps 16-byte alignment.
    __shared__ __align__(16) float As[2][16][KC + 4];

    const int rowTile = blockIdx.x * 16;
    const int tid  = threadIdx.x;
    const int wave = tid >> 5;        // 0..7  -> column tile
    const int lane = tid & 31;
    const int nBase = wave * 16;
    const int half = lane >> 4;       // 0: K pair {0,1}, 1: K pair {+2,+3}
    const int l15  = lane & 15;       // M index (A frag) / N index (B frag)

    // Staging map: 16 rows x KC floats = 256 b128 transfers, one per thread.
    const int ldRow = tid >> 4;       // 16 threads per row
    const int ldC4  = tid & 15;       // float4 slot within row (KC/4 = 16)
    const float* aBase = A + (size_t)(rowTile + ldRow) * NN + ldC4 * 4;

    // LDS byte addresses for this thread's b128 slot in each buffer
    // (flat->LDS: low 32 bits of the generic pointer are the LDS address).
    const unsigned ldsOff0 = (unsigned)(uintptr_t)&As[0][ldRow][ldC4 * 4];
    const unsigned ldsOff1 = (unsigned)(uintptr_t)&As[1][ldRow][ldC4 * 4];

    // Prologue: async-DMA chunk 0 into buffer 0.
    {
        unsigned long long ga = (unsigned long long)(uintptr_t)aBase;
        asm volatile("global_load_async_to_lds_b128 %0, %1, off"
                     :: "v"(ldsOff0), "v"(ga) : "memory");
    }

    v8f acc = {};

    for (int k0 = 0; k0 < NN; k0 += KC) {
        const int buf = (k0 / KC) & 1;

        // Wait for this wave's outstanding async DMA, then block-wide barrier
        // so every wave's slice of the current tile is resident in LDS.
        asm volatile("s_wait_asynccnt 0x0" ::: "memory");
        __syncthreads();

        // Kick off the DMA of the next chunk into the other buffer; it runs
        // concurrently with the WMMA chain below. (Uniform branch: EXEC
        // stays all-ones for the WMMAs.)
        if (k0 + KC < NN) {
            unsigned long long ga =
                (unsigned long long)(uintptr_t)(aBase + k0 + KC);
            unsigned dst = buf ? ldsOff0 : ldsOff1;
            asm volatile("global_load_async_to_lds_b128 %0, %1, off"
                         :: "v"(dst), "v"(ga) : "memory");
        }

#pragma unroll
        for (int kk = 0; kk < KC; kk += 4) {
            const int ka = kk + half * 2;
            v2f a, b;
            // A frag (16x4 f32): lanes 0-15 M=l15, VGPR0=K, VGPR1=K+1;
            // lanes 16-31 same M, K+2/K+3.
            a.x = As[buf][l15][ka];
            a.y = As[buf][l15][ka + 1];
            // B frag (4x16 f32): lanes 0-15 N=l15 rows K/K+1; lanes 16-31 K+2/K+3.
            const float* hp = Hs + (size_t)(k0 + ka) * DD + nBase + l15;
            b.x = hp[0];
            b.y = hp[DD];
            acc = __builtin_amdgcn_wmma_f32_16x16x4_f32(
                false, a, false, b, (short)0, acc, false, false);
        }
        // No trailing barrier needed: each wave's LDS reads of buf completed
        // before its WMMAs (DScnt), and buf is only overwritten after the
        // next iteration's barrier.
    }

    // D layout: VGPR v -> row = v + half*8, col = l15 (within tile)
#pragma unroll
    for (int v = 0; v < 8; ++v) {
        const int r = rowTile + v + half * 8;
        float val = acc[v] * dinv[r];
        val = val > 0.f ? val : 0.f;
        out[(size_t)r * DD + nBase + l15] = val;
    }
}

// ---------------------------------------------------------------------------
extern "C" void kernel_launch(void* const* d_in, const int* in_sizes, int n_in,
                              void* d_out, int out_size, void* d_ws, size_t ws_size,
                              hipStream_t stream) {
    const float* A = (const float*)d_in[0];   // [1, 8192, 8192]
    const float* X = (const float*)d_in[1];   // [1, 8192, 128]
    const float* W = (const float*)d_in[2];   // [128, 128]
    float* out = (float*)d_out;               // [1, 8192, 128]

    float* dinv = (float*)d_ws;               // 8192 floats
    float* Hs   = dinv + NN;                  // 8192*128 floats (4 MB)

    k_degree<<<NN, 256, 0, stream>>>(A, dinv);
    k_xw<<<NN, DD, 0, stream>>>(X, W, dinv, Hs);
    k_spmm<<<NN / 16, 256, 0, stream>>>(A, Hs, dinv, out);
}